// GCNTail_48936857370859
// MI455X (gfx1250) — compile-verified
//
#include <hip/hip_runtime.h>

#define HID 128
#define OUTD 64

typedef __attribute__((ext_vector_type(2))) float v2f;
typedef __attribute__((ext_vector_type(8))) float v8f;

// ---------------------------------------------------------------------------
// Degree = 1 (self loop) + count of incoming edges; integer atomics are exact.
// ---------------------------------------------------------------------------
__global__ __launch_bounds__(256) void gcn_init_deg(unsigned* __restrict__ deg, int n) {
  int i = blockIdx.x * 256 + threadIdx.x;
  if (i < n) deg[i] = 1u;
}

__global__ __launch_bounds__(256) void gcn_deg_accum(const long long* __restrict__ dst,
                                                     unsigned* __restrict__ deg, int e) {
  int i = blockIdx.x * 256 + threadIdx.x;
  if (i < e) atomicAdd(&deg[(int)dst[i]], 1u);
}

// In-place: reinterpret the uint degree as float rsqrt(deg).
__global__ __launch_bounds__(256) void gcn_dinv(float* __restrict__ dinv_io, int n) {
  int i = blockIdx.x * 256 + threadIdx.x;
  if (i < n) {
    unsigned d = ((const unsigned*)dinv_io)[i];
    dinv_io[i] = rsqrtf((float)d);
  }
}

// ---------------------------------------------------------------------------
// GEMM1: h = x @ W   (fp32 WMMA 16x16x4, one 16x16 tile per wave)
// A 16x4 layout: lanes 0-15 -> K=kk+0,kk+1 ; lanes 16-31 -> K=kk+2,kk+3
// C layout: VGPR r = row (r + 8*laneHalf), col = lane%16
// ---------------------------------------------------------------------------
__global__ __launch_bounds__(256) void gcn_gemm1(const float* __restrict__ x,
                                                 const float* __restrict__ W,
                                                 float* __restrict__ h, int ntiles) {
  const int tile = blockIdx.x * 8 + (threadIdx.x >> 5);
  if (tile >= ntiles) return;                 // wave-uniform, EXEC stays all-ones
  const int lane = threadIdx.x & 31;
  const int tm = tile >> 3;                   // HID/16 == 8 col-tiles
  const int tn = tile & 7;
  const int rowBase = tm * 16, colBase = tn * 16;
  const int half = lane >> 4;
  const int mn = lane & 15;
  const float* arow = x + (size_t)(rowBase + mn) * HID;

  v8f c = {0.f, 0.f, 0.f, 0.f, 0.f, 0.f, 0.f, 0.f};
  for (int kk = 0; kk < HID; kk += 4) {
    const int k0 = kk + half * 2;
    v2f a, bm;
    a.x = arow[k0];
    a.y = arow[k0 + 1];
    bm.x = W[(size_t)k0 * HID + colBase + mn];
    bm.y = W[(size_t)(k0 + 1) * HID + colBase + mn];
    c = __builtin_amdgcn_wmma_f32_16x16x4_f32(false, a, false, bm, (short)0, c,
                                              false, false);
  }
#pragma unroll
  for (int r = 0; r < 8; ++r)
    h[(size_t)(rowBase + half * 8 + r) * HID + colBase + mn] = c[r];
}

// ---------------------------------------------------------------------------
// Seed accumulator with the self-loop term: acc[i,f] = h[i,f] * dinv[i]^2
// ---------------------------------------------------------------------------
__global__ __launch_bounds__(256) void gcn_init_acc(const float* __restrict__ h,
                                                    const float* __restrict__ dinv,
                                                    float* __restrict__ acc, int total) {
  int i = blockIdx.x * 256 + threadIdx.x;
  if (i < total) {
    float di = dinv[i >> 7];   // /HID
    acc[i] = h[i] * di * di;
  }
}

// ---------------------------------------------------------------------------
// Edge scatter: thread = (edge, 4-feature chunk). float4 gather of h[src],
// hardware fp32 atomic adds into acc[dst]. This is the HBM-bound hot loop.
// ---------------------------------------------------------------------------
__global__ __launch_bounds__(256) void gcn_scatter(const long long* __restrict__ src,
                                                   const long long* __restrict__ dst,
                                                   const float* __restrict__ h,
                                                   const float* __restrict__ dinv,
                                                   float* __restrict__ acc, int e) {
  int gid = blockIdx.x * 256 + threadIdx.x;
  int edge = gid >> 5;
  if (edge >= e) return;
  int q = (gid & 31) * 4;
  int s = (int)src[edge];
  int d = (int)dst[edge];
  float coef = dinv[s] * dinv[d];
  const float4 hv = *(const float4*)(h + (size_t)s * HID + q);
  float* ap = acc + (size_t)d * HID + q;
  unsafeAtomicAdd(ap + 0, hv.x * coef);
  unsafeAtomicAdd(ap + 1, hv.y * coef);
  unsafeAtomicAdd(ap + 2, hv.z * coef);
  unsafeAtomicAdd(ap + 3, hv.w * coef);
}

// ---------------------------------------------------------------------------
// GEMM2: out = relu(acc + b) @ Wc + bc   (bias+ReLU fused into A load)
// ---------------------------------------------------------------------------
__global__ __launch_bounds__(256) void gcn_gemm2(const float* __restrict__ acc,
                                                 const float* __restrict__ bias,
                                                 const float* __restrict__ Wc,
                                                 const float* __restrict__ bc,
                                                 float* __restrict__ out, int ntiles) {
  const int tile = blockIdx.x * 8 + (threadIdx.x >> 5);
  if (tile >= ntiles) return;                 // wave-uniform
  const int lane = threadIdx.x & 31;
  const int tm = tile >> 2;                   // OUTD/16 == 4 col-tiles
  const int tn = tile & 3;
  const int rowBase = tm * 16, colBase = tn * 16;
  const int half = lane >> 4;
  const int mn = lane & 15;
  const float* arow = acc + (size_t)(rowBase + mn) * HID;

  v8f c = {0.f, 0.f, 0.f, 0.f, 0.f, 0.f, 0.f, 0.f};
  for (int kk = 0; kk < HID; kk += 4) {
    const int k0 = kk + half * 2;
    v2f a, bm;
    a.x = fmaxf(arow[k0] + bias[k0], 0.f);
    a.y = fmaxf(arow[k0 + 1] + bias[k0 + 1], 0.f);
    bm.x = Wc[(size_t)k0 * OUTD + colBase + mn];
    bm.y = Wc[(size_t)(k0 + 1) * OUTD + colBase + mn];
    c = __builtin_amdgcn_wmma_f32_16x16x4_f32(false, a, false, bm, (short)0, c,
                                              false, false);
  }
  const float add = bc[colBase + mn];
#pragma unroll
  for (int r = 0; r < 8; ++r)
    out[(size_t)(rowBase + half * 8 + r) * OUTD + colBase + mn] = c[r] + add;
}

// ---------------------------------------------------------------------------
extern "C" void kernel_launch(void* const* d_in, const int* in_sizes, int n_in,
                              void* d_out, int out_size, void* d_ws, size_t ws_size,
                              hipStream_t stream) {
  const float* x      = (const float*)d_in[0];
  const long long* ei = (const long long*)d_in[1];   // int64 [2, E]
  const float* W      = (const float*)d_in[2];
  const float* b      = (const float*)d_in[3];
  const float* Wc     = (const float*)d_in[4];
  const float* bc     = (const float*)d_in[5];
  float* out          = (float*)d_out;

  const int n = in_sizes[0] / HID;        // 100000 nodes
  const int e = in_sizes[1] / 2;          // 1600000 edges
  const long long* src = ei;
  const long long* dst = ei + e;

  // Workspace layout (floats): [deg/dinv: n][h: n*HID][acc: n*HID]
  float* ws   = (float*)d_ws;
  float* dinv = ws;                                   // uint deg -> float rsqrt in place
  float* h    = ws + n;                               // n*400000B offset: 16B aligned
  float* acc  = h + (size_t)n * HID;

  // 1) degrees (self-loop seeded)
  gcn_init_deg<<<(n + 255) / 256, 256, 0, stream>>>((unsigned*)dinv, n);
  gcn_deg_accum<<<(e + 255) / 256, 256, 0, stream>>>(dst, (unsigned*)dinv, e);
  gcn_dinv<<<(n + 255) / 256, 256, 0, stream>>>(dinv, n);

  // 2) h = x @ W  (fp32 WMMA)
  const int tiles1 = (n / 16) * (HID / 16);
  gcn_gemm1<<<(tiles1 + 7) / 8, 256, 0, stream>>>(x, W, h, tiles1);

  // 3) acc = self-loop term, then scatter-add edge messages
  const int total = n * HID;
  gcn_init_acc<<<(total + 255) / 256, 256, 0, stream>>>(h, dinv, acc, total);
  const long long sthreads = (long long)e * 32;
  gcn_scatter<<<(unsigned)((sthreads + 255) / 256), 256, 0, stream>>>(src, dst, h, dinv,
                                                                      acc, e);

  // 4) out = relu(acc + b) @ Wc + bc  (fp32 WMMA, fused epilogue)
  const int tiles2 = (n / 16) * (OUTD / 16);
  gcn_gemm2<<<(tiles2 + 7) / 8, 256, 0, stream>>>(acc, b, Wc, bc, out, tiles2);
}